// WeightGRU_27719718928963
// MI455X (gfx1250) — compile-verified
//
#include <hip/hip_runtime.h>
#include <math.h>

#define H      128      // hidden
#define STEPS  100
#define TILE   128      // rows/cols per block tile
#define LDSW   132      // padded LDS row stride (floats): banks = 4*m + k -> conflict-free

typedef float v2f __attribute__((ext_vector_type(2)));
typedef float v8f __attribute__((ext_vector_type(8)));
typedef unsigned int u32x4 __attribute__((ext_vector_type(4)));
typedef int i32x4 __attribute__((ext_vector_type(4)));
typedef int i32x8 __attribute__((ext_vector_type(8)));

// ---- Tensor Data Mover availability / arity probing -----------------------
#if defined(__has_builtin)
#if __has_builtin(__builtin_amdgcn_tensor_load_to_lds)
#define USE_TDM 1
#endif
#endif
#ifndef USE_TDM
#define USE_TDM 0
#endif

#if defined(__has_include)
#if __has_include(<hip/amd_detail/amd_gfx1250_TDM.h>)
#define TDM_ARITY6 1
#endif
#endif
#ifndef TDM_ARITY6
#define TDM_ARITY6 0
#endif

#if USE_TDM
// Issue a 2D TDM tile load: tile1 x tile0 elements (f32) from a row-major
// tensor (row stride stride0 elements) at gptr, into LDS at lds_off, with
// LDS padding of 4 DWORDs after every 128 DWORDs (row stride 132 floats).
// tdim0/tdim1 = valid extent measured from the tile origin (OOB reads -> 0).
__device__ __forceinline__ void tdm_load_tile(const void* gptr, unsigned lds_off,
                                              unsigned tdim0, unsigned tdim1,
                                              unsigned tile0, unsigned tile1,
                                              unsigned stride0) {
    const unsigned long long ga = (unsigned long long)gptr;
    u32x4 g0;
    g0[0] = 1u;                                    // count=1, user descriptor
    g0[1] = lds_off;                               // lds_addr (bytes)
    g0[2] = (unsigned)ga;                          // global_addr[31:0]
    g0[3] = ((unsigned)(ga >> 32) & 0x01FFFFFFu)   // global_addr[56:32]
          | (2u << 30);                            // type = 2 ("image")

    i32x8 g1;
    g1[0] = (int)((2u << 16)        // data_size = 4B
                | (1u << 20)        // pad_enable
                | (6u << 22)        // pad_interval: 128 DWORDs (512B = one row)
                | (3u << 25));      // pad_amount : 4 DWORDs  -> LDSW = 132
    g1[1] = (int)((tdim0 & 0xFFFFu) << 16);                        // tensor_dim0[15:0]
    g1[2] = (int)(((tdim0 >> 16) & 0xFFFFu) | ((tdim1 & 0xFFFFu) << 16));
    g1[3] = (int)(((tdim1 >> 16) & 0xFFFFu) | ((tile0 & 0xFFFFu) << 16));
    g1[4] = (int)(tile1 & 0xFFFFu);                                // tile_dim2 = 0
    g1[5] = (int)stride0;                                          // dim0 stride [31:0]
    g1[6] = 0;                                                     // stride hi / dim1 stride lo
    g1[7] = 0;

    i32x4 gz = {0, 0, 0, 0};
#if TDM_ARITY6
    i32x8 gz8 = {0, 0, 0, 0, 0, 0, 0, 0};
    __builtin_amdgcn_tensor_load_to_lds(g0, g1, gz, gz, gz8, 0);
#else
    __builtin_amdgcn_tensor_load_to_lds(g0, g1, gz, gz, 0);
#endif
}
#endif // USE_TDM

// ---------------------------------------------------------------------------
// init: ws_x = x0, ws_h = 0
// ---------------------------------------------------------------------------
__global__ void init_kernel(const float* __restrict__ x0,
                            float* __restrict__ ws_x,
                            float* __restrict__ ws_h, int D) {
    int i = blockIdx.x * blockDim.x + threadIdx.x;
    if (i < D) ws_x[i] = x0[i];
    if (blockIdx.x == 0 && threadIdx.x < H) ws_h[threadIdx.x] = 0.0f;
}

// ---------------------------------------------------------------------------
// Shared WMMA GEMV core: 8 waves, wave w owns rows 16w..16w+15 of the LDS
// tile; vector broadcast into every B column; result column 0 -> sout.
// ---------------------------------------------------------------------------
__device__ __forceinline__ void wmma_gemv_tile(const float* sW, const float* sv,
                                               float* sout, int tid) {
    const int lane = tid & 31;
    const int w    = tid >> 5;
    const int m    = (w << 4) + (lane & 15);   // A-matrix row for this lane
    const int hi   = (lane >> 4) << 1;         // lanes 16-31 hold K=2,3

    v8f acc = {};
    #pragma unroll
    for (int k = 0; k < TILE; k += 4) {
        const int col = k + hi;
        v2f a, b;
        a.x = sW[m * LDSW + col];
        a.y = sW[m * LDSW + col + 1];
        b.x = sv[col];
        b.y = sv[col + 1];
        acc = __builtin_amdgcn_wmma_f32_16x16x4_f32(
                  false, a, false, b, (short)0, acc, false, false);
    }
    // D layout: VGPR r, lane0 -> (M=r, N=0); lane16 -> (M=8+r, N=0)
    if ((lane & 15) == 0) {
        const int mbase = (w << 4) + ((lane >> 4) << 3);
        #pragma unroll
        for (int r = 0; r < 8; ++r) sout[mbase + r] = acc[r];
    }
}

// ---------------------------------------------------------------------------
// encode: partial[b][m] = sum_{d in chunk b} W_enc[m][d] * x[d]
// ---------------------------------------------------------------------------
__global__ __launch_bounds__(256) void enc_kernel(const float* __restrict__ Wenc,
                                                  const float* __restrict__ x,
                                                  float* __restrict__ part, int D) {
    extern __shared__ float smem[];
    float* sW   = smem;                 // H * LDSW
    float* sx   = smem + H * LDSW;      // LDSW
    float* sout = sx + LDSW;            // H

    const int tid = threadIdx.x;
    const int d0  = blockIdx.x * TILE;

#if USE_TDM
    if (tid < 32) {   // one wave issues the DMA: 128 rows x 128 cols, stride D
        tdm_load_tile(Wenc + d0, (unsigned)(unsigned long long)sW,
                      (unsigned)(D - d0), H, TILE, H, (unsigned)D);
    }
#else
    for (int v = tid; v < (H * TILE) / 4; v += 256) {
        int i = v << 2, r = i >> 7, c = i & (TILE - 1);
        int d = d0 + c;
        float4 val;
        if (d + 3 < D) {
            val = *(const float4*)(Wenc + (size_t)r * D + d);
        } else {
            val.x = (d     < D) ? Wenc[(size_t)r * D + d]     : 0.0f;
            val.y = (d + 1 < D) ? Wenc[(size_t)r * D + d + 1] : 0.0f;
            val.z = (d + 2 < D) ? Wenc[(size_t)r * D + d + 2] : 0.0f;
            val.w = (d + 3 < D) ? Wenc[(size_t)r * D + d + 3] : 0.0f;
        }
        *(float4*)(sW + r * LDSW + c) = val;
    }
#endif
    if (tid < TILE) { int d = d0 + tid; sx[tid] = (d < D) ? x[d] : 0.0f; }
#if USE_TDM
    if (tid < 32) __builtin_amdgcn_s_wait_tensorcnt(0);
#endif
    __syncthreads();

    wmma_gemv_tile(sW, sx, sout, tid);
    __syncthreads();
    if (tid < H) part[(size_t)blockIdx.x * H + tid] = sout[tid];
}

// ---------------------------------------------------------------------------
// GRU cell: deterministic partial reduce, gates, h update (tiny, VALU)
// ---------------------------------------------------------------------------
__global__ __launch_bounds__(384) void gru_kernel(const float* __restrict__ part,
                                                  const float* __restrict__ benc,
                                                  const float* __restrict__ Wih,
                                                  const float* __restrict__ Whh,
                                                  const float* __restrict__ bih,
                                                  const float* __restrict__ bn,
                                                  float* __restrict__ hstate, int NB) {
    __shared__ float s_emb[H], s_h[H], s_ag[3 * H], s_bg[3 * H];
    const int tid = threadIdx.x;

    if (tid < H) {
        float sum = benc[tid];
        for (int b = 0; b < NB; ++b) sum += part[(size_t)b * H + tid];
        s_emb[tid] = sum;
        s_h[tid]   = hstate[tid];
    }
    __syncthreads();

    {   // ag = W_ih @ emb + b_ih ; bg = W_hh @ h
        float ag = bih[tid], bg = 0.0f;
        const float* wi = Wih + (size_t)tid * H;
        const float* wh = Whh + (size_t)tid * H;
        #pragma unroll 8
        for (int k = 0; k < H; ++k) {
            ag = fmaf(wi[k], s_emb[k], ag);
            bg = fmaf(wh[k], s_h[k],   bg);
        }
        s_ag[tid] = ag; s_bg[tid] = bg;
    }
    __syncthreads();

    if (tid < H) {
        float r  = 1.0f / (1.0f + expf(-(s_ag[tid]       + s_bg[tid])));
        float ii = 1.0f / (1.0f + expf(-(s_ag[tid + H]   + s_bg[tid + H])));
        float nn = tanhf(s_ag[tid + 2 * H] + r * (s_bg[tid + 2 * H] + bn[tid]));
        hstate[tid] = nn + ii * (s_h[tid] - nn);
    }
}

// ---------------------------------------------------------------------------
// decode: x[row] = W_dec[row]·h + b_dec[row]; writes samples/means/sigmas
// ---------------------------------------------------------------------------
__global__ __launch_bounds__(256) void dec_kernel(const float* __restrict__ Wdec,
                                                  const float* __restrict__ hvec,
                                                  const float* __restrict__ bdec,
                                                  float* __restrict__ xnext,
                                                  float* __restrict__ outS,
                                                  float* __restrict__ outM,
                                                  float* __restrict__ outSig, int D) {
    extern __shared__ float smem[];
    float* sW   = smem;
    float* sh   = smem + H * LDSW;
    float* sout = sh + LDSW;

    const int tid = threadIdx.x;
    const int r0  = blockIdx.x * TILE;

#if USE_TDM
    if (tid < 32) {   // contiguous 64KB tile: 128 rows x 128 cols, stride H
        tdm_load_tile(Wdec + (size_t)r0 * H, (unsigned)(unsigned long long)sW,
                      H, (unsigned)(D - r0), H, TILE, H);
    }
#else
    for (int v = tid; v < (H * TILE) / 4; v += 256) {
        int i = v << 2, r = i >> 7, c = i & (TILE - 1);
        int row = r0 + r;
        float4 val = {0.0f, 0.0f, 0.0f, 0.0f};
        if (row < D) val = *(const float4*)(Wdec + (size_t)row * H + c);
        *(float4*)(sW + r * LDSW + c) = val;
    }
#endif
    if (tid < H) sh[tid] = hvec[tid];
#if USE_TDM
    if (tid < 32) __builtin_amdgcn_s_wait_tensorcnt(0);
#endif
    __syncthreads();

    wmma_gemv_tile(sW, sh, sout, tid);
    __syncthreads();

    if (tid < TILE) {
        int row = r0 + tid;
        if (row < D) {
            float v = sout[tid] + bdec[row];
            xnext[row]  = v;      // feeds next step's encode
            outS[row]   = v;      // samples
            outM[row]   = v;      // means
            outSig[row] = 0.0f;   // sigmas
        }
    }
}

// ---------------------------------------------------------------------------
extern "C" void kernel_launch(void* const* d_in, const int* in_sizes, int n_in,
                              void* d_out, int out_size, void* d_ws, size_t ws_size,
                              hipStream_t stream) {
    const float* x0   = (const float*)d_in[0];
    const float* Wenc = (const float*)d_in[1];
    const float* benc = (const float*)d_in[2];
    const float* Wih  = (const float*)d_in[3];
    const float* Whh  = (const float*)d_in[4];
    const float* bih  = (const float*)d_in[5];
    const float* bn   = (const float*)d_in[6];
    const float* Wdec = (const float*)d_in[7];
    const float* bdec = (const float*)d_in[8];

    const int D  = in_sizes[0];                 // 50000
    const int NB = (D + TILE - 1) / TILE;       // 391

    float* out     = (float*)d_out;
    float* wsf     = (float*)d_ws;
    float* ws_h    = wsf;                       // H
    float* ws_x    = wsf + H;                   // NB*TILE (>= D)
    float* ws_part = ws_x + (size_t)NB * TILE;  // NB*H

    const size_t SM   = (size_t)STEPS * D;
    const size_t smem = (size_t)(H * LDSW + LDSW + H) * sizeof(float); // ~68.6 KB of 320 KB

    init_kernel<<<(D + 255) / 256, 256, 0, stream>>>(x0, ws_x, ws_h, D);

    for (int s = 0; s < STEPS; ++s) {
        enc_kernel<<<NB, 256, smem, stream>>>(Wenc, ws_x, ws_part, D);
        gru_kernel<<<1, 384, 0, stream>>>(ws_part, benc, Wih, Whh, bih, bn, ws_h, NB);
        dec_kernel<<<NB, 256, smem, stream>>>(Wdec, ws_h, bdec, ws_x,
                                              out + (size_t)s * D,
                                              out + SM + (size_t)s * D,
                                              out + 2 * SM + (size_t)s * D, D);
    }
}